// Random_61452392071527
// MI455X (gfx1250) — compile-verified
//
#include <hip/hip_runtime.h>
#include <math.h>

// ---------------------------------------------------------------------------
// Reference: z = c*mu + sqrt(1-c^2) * (noise * sqrt(var_row(mu))),  c = 1.0
// With c == 1.0 the noise weight is exactly 0.0f and mu/noise are finite
// (standard normals), so z == mu bit-exactly. The optimal MI455X kernel is a
// 512 MiB streaming copy (~22.5 us at 23.3 TB/s). We implement it with the
// CDNA5 async global<->LDS mover (ASYNCcnt-tracked, no VGPR staging,
// per-wave overlap of store(k) with load(k+1)). A general kernel handling
// any c (row variance reduction + combine) is also provided.
// ---------------------------------------------------------------------------

#ifndef __has_builtin
#define __has_builtin(x) 0
#endif

#if __has_builtin(__builtin_amdgcn_global_load_async_to_lds_b128) && \
    __has_builtin(__builtin_amdgcn_global_store_async_from_lds_b128)
#define HAVE_ASYNC_LDS 1
#else
#define HAVE_ASYNC_LDS 0
#endif

// Builtin parameter types (from hipcc diagnostics): int4-vector pointers,
// global side in addrspace(1) ("__device__"), LDS side in addrspace(3).
typedef int v4i __attribute__((vector_size(16)));
typedef v4i __attribute__((address_space(1)))* gptr_t;  // 64-bit global ptr
typedef v4i __attribute__((address_space(3)))* lptr_t;  // 32-bit LDS ptr

// AS0(generic) and AS1 share address values on amdgcn -> exact reinterpret.
__device__ __forceinline__ gptr_t as_global(const void* p) {
    return (gptr_t)(uintptr_t)p;
}
// Low 32 bits of a generic LDS pointer are the LDS byte address (ISA 10.2);
// AS3 pointers are 32-bit.
__device__ __forceinline__ lptr_t as_lds(const void* p) {
    return (lptr_t)(unsigned)(uintptr_t)p;
}

__device__ __forceinline__ void wait_async0() {
#if __has_builtin(__builtin_amdgcn_s_wait_asynccnt)
    __builtin_amdgcn_s_wait_asynccnt(0);
#else
    asm volatile("s_wait_asynccnt 0" ::: "memory");
#endif
}

// ------------------------- streaming copy (c == 1) -------------------------

constexpr int TPB       = 256;                 // 8 waves (wave32)
constexpr int OPS       = 8;                   // b128 async ops / thread / chunk
constexpr int VEC       = 16;                  // bytes per b128 op
constexpr int CHUNK     = TPB * OPS * VEC;     // 32 KiB per chunk
constexpr int LDS_BYTES = 2 * CHUNK;           // 64 KiB double buffer

#if HAVE_ASYNC_LDS
__global__ __launch_bounds__(TPB)
void stream_copy_async(const char* __restrict__ src, char* __restrict__ dst,
                       size_t nchunks) {
    __shared__ char smem[LDS_BYTES];
    const int t = threadIdx.x;
    size_t c = blockIdx.x;                 // uniform per block
    const size_t stride = gridDim.x;
    int buf = 0;

    if (c < nchunks) {
        const size_t base = c * (size_t)CHUNK;
        #pragma unroll
        for (int j = 0; j < OPS; ++j) {
            const size_t off = (size_t)(j * TPB + t) * VEC;   // coalesced 512B/wave
            __builtin_amdgcn_global_load_async_to_lds_b128(
                as_global(src + base + off), as_lds(smem + buf * CHUNK + off), 0, 0);
        }
    }
    while (c < nchunks) {
        const size_t nc = c + stride;
        // All my outstanding async ops done: loads L(c) into buf, and the
        // previous iteration's stores out of buf^1 (so buf^1 is reusable).
        wait_async0();
        const size_t base = c * (size_t)CHUNK;
        #pragma unroll
        for (int j = 0; j < OPS; ++j) {
            const size_t off = (size_t)(j * TPB + t) * VEC;
            __builtin_amdgcn_global_store_async_from_lds_b128(
                as_global(dst + base + off), as_lds(smem + buf * CHUNK + off), 0, 0);
        }
        if (nc < nchunks) {                 // prefetch next chunk, overlaps stores
            const size_t nbase = nc * (size_t)CHUNK;
            #pragma unroll
            for (int j = 0; j < OPS; ++j) {
                const size_t off = (size_t)(j * TPB + t) * VEC;
                __builtin_amdgcn_global_load_async_to_lds_b128(
                    as_global(src + nbase + off),
                    as_lds(smem + (buf ^ 1) * CHUNK + off), 0, 0);
            }
        }
        buf ^= 1;
        c = nc;
    }
    // s_endpgm performs an implicit wait-idle, draining the final stores.
}
#endif

__global__ __launch_bounds__(TPB)
void stream_copy_vec4(const float4* __restrict__ src, float4* __restrict__ dst,
                      size_t n4) {
    size_t i = (size_t)blockIdx.x * blockDim.x + threadIdx.x;
    const size_t stride = (size_t)gridDim.x * blockDim.x;
    for (; i < n4; i += stride) dst[i] = src[i];
}

__global__ void stream_copy_scalar(const float* __restrict__ src,
                                   float* __restrict__ dst, size_t n) {
    size_t i = (size_t)blockIdx.x * blockDim.x + threadIdx.x;
    if (i < n) dst[i] = src[i];
}

// --------------------- general path (any c, unused for c==1) ----------------

__global__ __launch_bounds__(256)
void corr_combine_general(const float* __restrict__ mu,
                          const float* __restrict__ noise,
                          float* __restrict__ z,
                          int D, float c, float w /* = sqrt(1-c^2) */) {
    const size_t row = blockIdx.x;
    const float* m = mu    + row * (size_t)D;
    const float* x = noise + row * (size_t)D;
    float*       o = z     + row * (size_t)D;

    float s = 0.f, s2 = 0.f;
    for (int i = threadIdx.x; i < D; i += blockDim.x) {
        const float v = m[i];
        s += v; s2 = fmaf(v, v, s2);
    }
    __shared__ float rs[256], rq[256];
    rs[threadIdx.x] = s; rq[threadIdx.x] = s2;
    __syncthreads();
    for (int off = 128; off > 0; off >>= 1) {
        if (threadIdx.x < off) {
            rs[threadIdx.x] += rs[threadIdx.x + off];
            rq[threadIdx.x] += rq[threadIdx.x + off];
        }
        __syncthreads();
    }
    const float invD = 1.0f / (float)D;
    const float mean = rs[0] * invD;
    const float var  = fmaxf(rq[0] * invD - mean * mean, 0.0f);
    const float ws   = w * sqrtf(var);
    for (int i = threadIdx.x; i < D; i += blockDim.x)
        o[i] = fmaf(c, m[i], ws * x[i]);
}

// ------------------------------- launch -------------------------------------

extern "C" void kernel_launch(void* const* d_in, const int* in_sizes, int n_in,
                              void* d_out, int out_size, void* d_ws, size_t ws_size,
                              hipStream_t stream) {
    const float* mu    = (const float*)d_in[0];
    const float* noise = (const float*)d_in[1];
    float*       out   = (float*)d_out;
    const size_t n     = (size_t)out_size;          // B*D = 67,108,864 fp32

    constexpr float kC = 1.0f;                      // TARGET_CORRELATION
    constexpr float kW = 0.0f;                      // sqrt(1 - kC*kC) == 0 exactly
    constexpr bool  kIdentity = (kW == 0.0f);

    if (kIdentity) {
        const size_t nbytes  = n * sizeof(float);
        const size_t nchunks = nbytes / (size_t)CHUNK;
#if HAVE_ASYNC_LDS
        if (nchunks) {
            // 64 KiB LDS/block -> ~5 blocks/WGP resident; 2048 blocks covers
            // the device several times over; 8192 chunks total for 256 MiB.
            unsigned grid = (unsigned)((nchunks < 2048) ? nchunks : 2048);
            stream_copy_async<<<grid, TPB, 0, stream>>>((const char*)mu,
                                                        (char*)out, nchunks);
        }
#else
        if (nchunks) {
            const size_t n4 = (nchunks * (size_t)CHUNK) / sizeof(float4);
            stream_copy_vec4<<<2048, TPB, 0, stream>>>((const float4*)mu,
                                                       (float4*)out, n4);
        }
#endif
        // Tail (none for 256 MiB, but stay generic).
        const size_t done = nchunks * (size_t)CHUNK / sizeof(float);
        const size_t rem  = n - done;
        if (rem)
            stream_copy_scalar<<<(unsigned)((rem + 255) / 256), 256, 0, stream>>>(
                mu + done, out + done, rem);
    } else {
        const int D = 4096;
        const unsigned rows = (unsigned)(n / (size_t)D);
        corr_combine_general<<<rows, 256, 0, stream>>>(mu, noise, out, D, kC, kW);
    }
}